// ModelNew_43173011259812
// MI455X (gfx1250) — compile-verified
//
#include <hip/hip_runtime.h>
#include <hip/hip_bf16.h>

typedef __attribute__((ext_vector_type(2))) float v2f;
typedef __attribute__((ext_vector_type(8))) float v8f;

#define S_LEN 4096
#define H_DIM 16
#define B_DIM 2
#define P_DIM 64
#define N_DIM 128
#define BH (B_DIM * H_DIM)             // 32
#define OUT_ELEMS (BH * P_DIM * N_DIM) // 262144

// ---------------------------------------------------------------------------
// Kernel 1: decay weights.  W[bh][t] = exp( sum_{t' > t} A[b, t', h] )
// One block per (b,h). Fixed-order block scan -> deterministic fp32 result.
// ---------------------------------------------------------------------------
__global__ void k_decay(const float* __restrict__ A, float* __restrict__ W) {
    __shared__ float ssum[256];
    __shared__ float stot;
    const int bh  = blockIdx.x;
    const int b   = bh >> 4;
    const int h   = bh & 15;
    const int tid = threadIdx.x;
    const int PER = S_LEN / 256;      // 16 timesteps per thread

    float loc[PER];
    float s = 0.f;
    const int tbase = tid * PER;
    for (int j = 0; j < PER; ++j) {
        loc[j] = A[(size_t)(b * S_LEN + tbase + j) * H_DIM + h];
        s += loc[j];
    }
    ssum[tid] = s;
    __syncthreads();
    if (tid == 0) {
        float run = 0.f;
        for (int i = 0; i < 256; ++i) { float v = ssum[i]; ssum[i] = run; run += v; }
        stot = run;
    }
    __syncthreads();
    float run = ssum[tid];
    const float tot = stot;
    for (int j = 0; j < PER; ++j) {
        run += loc[j];                                  // inclusive cumsum through t
        W[(size_t)bh * S_LEN + tbase + j] = expf(tot - run);
    }
}

// ---------------------------------------------------------------------------
// Kernel 2: split-K WMMA GEMM.
//   part[split][bh][p][n] = sum_{t in split} W[bh][t] * X[b,t,h,p] * B[b,t,h,n]
// 128 threads = 4 waves. Wave tile: 32 (p) x 64 (n) = 2 M-tiles x 4 N-tiles
// -> 8 v8f accumulators, 8 WMMAs per K-step of 4 timesteps, B tiles shared
// across both M-tiles (1.75 loads per WMMA).
// fp32 WMMA 16x16x4 layout: lanes 0-15 carry K=0,1 ; lanes 16-31 carry K=2,3.
// ---------------------------------------------------------------------------
template <int TPER>
__global__ void k_chunk_gemm(const float* __restrict__ X,
                             const float* __restrict__ Bm,
                             const float* __restrict__ W,
                             float* __restrict__ out) {
    constexpr int SPLITS = S_LEN / TPER;
    const int blk   = blockIdx.x;
    const int split = blk % SPLITS;
    const int bh    = blk / SPLITS;
    const int b     = bh >> 4;
    const int h     = bh & 15;
    const int t0    = split * TPER;

    const int tid   = threadIdx.x;
    const int wave  = tid >> 5;
    const int lane  = tid & 31;
    const int half  = lane >> 4;      // K half of the wave
    const int lm    = lane & 15;      // M (for A) / N (for B) index
    const int mpair = wave & 1;       // p rows [0,32) or [32,64)
    const int nhalf = wave >> 1;      // n cols [0,64) or [64,128)

    v8f acc[2][4] = {};

    const int p0 = mpair * 32 + lm;   // first M-tile row for this lane
    const int nb = nhalf * 64 + lm;   // first N-tile col for this lane

    const float* wrow = W + (size_t)bh * S_LEN;
    const long xstep = (long)H_DIM * P_DIM;   // X stride per timestep (1024)
    const long bstep = (long)H_DIM * N_DIM;   // B stride per timestep (2048)
    const long xbase = ((long)(b * S_LEN + t0) * H_DIM + h) * P_DIM + p0;
    const long bbase = ((long)(b * S_LEN + t0) * H_DIM + h) * N_DIM + nb;

    for (int k = 0; k < TPER; k += 4) {
        const int kk = k + half * 2;          // this half-wave's first K slot
        const int tE = t0 + kk;
        const float wE = wrow[tE];
        const float wO = wrow[tE + 1];

        const long xo = xbase + (long)kk * xstep;
        v2f a0, a1;                            // two 16-row M strips
        a0.x = wE * X[xo];
        a0.y = wO * X[xo + xstep];
        a1.x = wE * X[xo + 16];
        a1.y = wO * X[xo + xstep + 16];

        const long bo = bbase + (long)kk * bstep;
        v2f bt[4];
#pragma unroll
        for (int j = 0; j < 4; ++j) {
            bt[j].x = Bm[bo + j * 16];
            bt[j].y = Bm[bo + bstep + j * 16];
        }
#pragma unroll
        for (int j = 0; j < 4; ++j)
            acc[0][j] = __builtin_amdgcn_wmma_f32_16x16x4_f32(
                false, a0, false, bt[j], (short)0, acc[0][j], false, false);
#pragma unroll
        for (int j = 0; j < 4; ++j)
            acc[1][j] = __builtin_amdgcn_wmma_f32_16x16x4_f32(
                false, a1, false, bt[j], (short)0, acc[1][j], false, false);
    }

    // C/D layout: VGPR r, lanes 0-15 -> M=r ; lanes 16-31 -> M=r+8 ; N = lm
    float* obase = out + (long)split * OUT_ELEMS + (long)bh * (P_DIM * N_DIM);
    const int ncol = nhalf * 64 + lm;
#pragma unroll
    for (int i = 0; i < 2; ++i) {
        const int prow = mpair * 32 + i * 16 + 8 * half;
#pragma unroll
        for (int r = 0; r < 8; ++r) {
            const long o = (long)(prow + r) * N_DIM + ncol;
#pragma unroll
            for (int j = 0; j < 4; ++j)
                obase[o + j * 16] = acc[i][j][r];
        }
    }
}

// ---------------------------------------------------------------------------
// Kernel 3: fixed-order reduction of split partials -> deterministic.
// ---------------------------------------------------------------------------
__global__ void k_reduce(const float* __restrict__ part, float* __restrict__ out,
                         int splits) {
    const int i = blockIdx.x * blockDim.x + threadIdx.x;
    float s = 0.f;
    for (int p = 0; p < splits; ++p) s += part[(long)p * OUT_ELEMS + i];
    out[i] = s;
}

// ---------------------------------------------------------------------------
extern "C" void kernel_launch(void* const* d_in, const int* in_sizes, int n_in,
                              void* d_out, int out_size, void* d_ws, size_t ws_size,
                              hipStream_t stream) {
    const float* X = (const float*)d_in[0];
    const float* A = (const float*)d_in[1];
    const float* B = (const float*)d_in[2];
    // d_in[3] (C) is unused by the reference.
    float* out = (float*)d_out;

    float* W = (float*)d_ws;                                   // 512 KB
    const size_t needW = (size_t)BH * S_LEN * sizeof(float);
    const size_t per   = (size_t)OUT_ELEMS * sizeof(float);    // 1 MB / split

    int splits = 1;
    bool direct = true;
    if (ws_size >= needW + per) {
        const size_t avail = (ws_size - needW) / per;
        splits = (avail >= 16) ? 16 : (avail >= 8) ? 8 : (avail >= 4) ? 4
               : (avail >= 2) ? 2 : 1;
        direct = false;
    }

    k_decay<<<BH, 256, 0, stream>>>(A, W);

    float* part = direct ? out : (float*)((char*)d_ws + needW);
    switch (splits) {
        case 16: k_chunk_gemm< 256><<<BH * 16, 128, 0, stream>>>(X, B, W, part); break;
        case  8: k_chunk_gemm< 512><<<BH *  8, 128, 0, stream>>>(X, B, W, part); break;
        case  4: k_chunk_gemm<1024><<<BH *  4, 128, 0, stream>>>(X, B, W, part); break;
        case  2: k_chunk_gemm<2048><<<BH *  2, 128, 0, stream>>>(X, B, W, part); break;
        default: k_chunk_gemm<4096><<<BH *  1, 128, 0, stream>>>(X, B, W, part); break;
    }

    if (!direct) {
        k_reduce<<<OUT_ELEMS / 256, 256, 0, stream>>>(part, out, splits);
    }
}